// Sampler_1632087573248
// MI455X (gfx1250) — compile-verified
//
#include <hip/hip_runtime.h>
#include <math.h>

// -------------------------------------------------------------------------
// Gumbel-max (exponential-race) sampler for gfx1250.
// argmax_v softmax(l/t)_v/(E_v+eps)  ==  argmax_v  l_v/t - log(E_v+eps)
// (monotone per-row transform), greedy argmax(l) when t <= 0.
// Bandwidth-bound: 262 MB of logits streamed once via the CDNA5 async
// global->LDS engine (ASYNCcnt pipeline, 8 chunks in flight per wave).
// HBM roofline @ 23.3 TB/s: ~11.3 us.
// -------------------------------------------------------------------------

#define SAMPLER_EPS 1e-10f

constexpr int BLOCK = 256;   // 8 waves (wave32)
constexpr int DEPTH = 8;     // async pipeline depth (per-wave in-flight loads)

typedef int v4i __attribute__((ext_vector_type(4)));
typedef __attribute__((address_space(1))) v4i glob_v4i;   // HIP __device__ (AS1)
typedef __attribute__((address_space(3))) v4i lds_v4i;    // LDS (AS3)

#if defined(__has_builtin)
#if __has_builtin(__builtin_amdgcn_global_load_async_to_lds_b128)
#define HAVE_ASYNC_BUILTIN 1
#endif
#if __has_builtin(__builtin_amdgcn_s_wait_asynccnt)
#define HAVE_ASYNCCNT_BUILTIN 1
#endif
#endif

__device__ __forceinline__ void async_cp_b128(const float4* gsrc, float4* lds_dst) {
#if defined(HAVE_ASYNC_BUILTIN)
    __builtin_amdgcn_global_load_async_to_lds_b128(
        (glob_v4i*)(size_t)gsrc, (lds_v4i*)lds_dst, 0, 0);
#else
    unsigned int loff =
        (unsigned int)(size_t)(__attribute__((address_space(3))) void*)lds_dst;
    unsigned long long gaddr = (unsigned long long)(size_t)gsrc;
    asm volatile("global_load_async_to_lds_b128 %0, %1, off"
                 :: "v"(loff), "v"(gaddr) : "memory");
#endif
}

__device__ __forceinline__ void wait_asynccnt_le7() {
#if defined(HAVE_ASYNCCNT_BUILTIN)
    __builtin_amdgcn_s_wait_asynccnt(DEPTH - 1);
#else
    asm volatile("s_wait_asynccnt 7" ::: "memory");
#endif
}

__device__ __forceinline__ void wait_asynccnt_0() {
#if defined(HAVE_ASYNCCNT_BUILTIN)
    __builtin_amdgcn_s_wait_asynccnt(0);
#else
    asm volatile("s_wait_asynccnt 0" ::: "memory");
#endif
}

__device__ __forceinline__ void wait_dscnt0() {
    asm volatile("s_wait_dscnt 0" ::: "memory");
}

__device__ __forceinline__ void upd(float k, int idx, float& best, int& bidx) {
    if (k > best) { best = k; bidx = idx; }   // strict > keeps earliest index
}

__device__ __forceinline__ void proc4(float4 lv, float4 gv, float invT, float gs,
                                      int base, float& best, int& bidx) {
    upd(fmaf(lv.x, invT, gs * gv.x), base + 0, best, bidx);
    upd(fmaf(lv.y, invT, gs * gv.y), base + 1, best, bidx);
    upd(fmaf(lv.z, invT, gs * gv.z), base + 2, best, bidx);
    upd(fmaf(lv.w, invT, gs * gv.w), base + 3, best, bidx);
}

// ---- Main kernel: one block per token row; fused key + argmax ----------
__global__ void __launch_bounds__(BLOCK)
sampler_kernel(const float* __restrict__ logits,
               const float* __restrict__ temps,
               const float* __restrict__ gsrc,   // g[] if precomputed, else E[]
               int* __restrict__ out, int V, int need_log) {
    const int row = blockIdx.x;
    const int tid = threadIdx.x;

    const float temp   = temps[row];
    const bool  greedy = (temp <= 0.0f);
    // v_rcp_f32 (~1 ulp) is ample precision for an argmax key.
    const float invT   = greedy ? 1.0f
                                : __builtin_amdgcn_rcpf(fmaxf(temp, SAMPLER_EPS));
    const float gs     = greedy ? 0.0f : 1.0f;

    const float4* rowp = (const float4*)(logits + (size_t)row * (size_t)V);
    const float4* gp   = (const float4*)gsrc;
    const int nv4   = V >> 2;
    const int iters = nv4 / BLOCK;

    float best = -__builtin_inff();
    int   bidx = 0;

    __shared__ float4 buf[DEPTH][BLOCK];

    // Prologue: fill the async pipeline (each lane owns cell [slot][tid]).
    const int pro = (iters < DEPTH) ? iters : DEPTH;
    for (int j = 0; j < pro; ++j)
        async_cp_b128(rowp + (size_t)j * BLOCK + tid, &buf[j][tid]);

    // Steady state: exactly DEPTH loads in flight, so asynccnt<=DEPTH-1
    // guarantees (in-order completion) that chunk j has landed in LDS.
    const int main_iters = (iters > DEPTH) ? (iters - DEPTH) : 0;
    for (int j = 0; j < main_iters; ++j) {
        const int slot = j % DEPTH;
        wait_asynccnt_le7();                     // oldest chunk landed in LDS
        float4 lv = buf[slot][tid];              // ds_load of our own cell
        float4 gv = gp[(size_t)j * BLOCK + tid]; // L2-resident Gumbel terms
        if (need_log) {
            gv.x = -logf(gv.x + SAMPLER_EPS); gv.y = -logf(gv.y + SAMPLER_EPS);
            gv.z = -logf(gv.z + SAMPLER_EPS); gv.w = -logf(gv.w + SAMPLER_EPS);
        }
        wait_dscnt0();                           // LDS read retired before reuse
        async_cp_b128(rowp + (size_t)(j + DEPTH) * BLOCK + tid, &buf[slot][tid]);
        proc4(lv, gv, invT, gs, 4 * (j * BLOCK + tid), best, bidx);
    }

    // Epilogue: drain once, then the last <=DEPTH slots need no waits.
    wait_asynccnt_0();
    for (int j = main_iters; j < iters; ++j) {
        float4 lv = buf[j % DEPTH][tid];
        float4 gv = gp[(size_t)j * BLOCK + tid];
        if (need_log) {
            gv.x = -logf(gv.x + SAMPLER_EPS); gv.y = -logf(gv.y + SAMPLER_EPS);
            gv.z = -logf(gv.z + SAMPLER_EPS); gv.w = -logf(gv.w + SAMPLER_EPS);
        }
        proc4(lv, gv, invT, gs, 4 * (j * BLOCK + tid), best, bidx);
    }

    // float4 tail (none for V=128000, kept for generality)
    const int tb = iters * BLOCK;
    if (tb + tid < nv4) {
        float4 lv = rowp[tb + tid];
        float4 gv = gp[tb + tid];
        if (need_log) {
            gv.x = -logf(gv.x + SAMPLER_EPS); gv.y = -logf(gv.y + SAMPLER_EPS);
            gv.z = -logf(gv.z + SAMPLER_EPS); gv.w = -logf(gv.w + SAMPLER_EPS);
        }
        proc4(lv, gv, invT, gs, 4 * (tb + tid), best, bidx);
    }
    // scalar tail
    for (int i = (nv4 << 2) + tid; i < V; i += BLOCK) {
        float gvv = gsrc[i];
        if (need_log) gvv = -logf(gvv + SAMPLER_EPS);
        upd(fmaf(logits[(size_t)row * V + i], invT, gs * gvv), i, best, bidx);
    }

    // Block-wide argmax with lowest-index tie-break (matches jnp.argmax).
    __shared__ float sval[BLOCK];
    __shared__ int   sidx[BLOCK];
    sval[tid] = best; sidx[tid] = bidx;
    __syncthreads();
    for (int s = BLOCK / 2; s > 0; s >>= 1) {
        if (tid < s) {
            float ov = sval[tid + s]; int oi = sidx[tid + s];
            if (ov > sval[tid] || (ov == sval[tid] && oi < sidx[tid])) {
                sval[tid] = ov; sidx[tid] = oi;
            }
        }
        __syncthreads();
    }
    if (tid == 0) out[row] = sidx[0];
}

// ---- Prep kernel: g[v] = -log(E[v] + eps) (512 KB, L2-resident) --------
__global__ void gumbel_prep_kernel(const float* __restrict__ E,
                                   float* __restrict__ g, int V) {
    int i = blockIdx.x * blockDim.x + threadIdx.x;
    if (i < V) g[i] = -logf(E[i] + SAMPLER_EPS);
}

extern "C" void kernel_launch(void* const* d_in, const int* in_sizes, int n_in,
                              void* d_out, int out_size, void* d_ws, size_t ws_size,
                              hipStream_t stream) {
    const float* logits = (const float*)d_in[0];
    const float* temps  = (const float*)d_in[1];
    const float* expn   = (const float*)d_in[2];
    int*         out    = (int*)d_out;

    const int T = in_sizes[1];   // 512 tokens
    const int V = in_sizes[2];   // 128000 vocab

    const bool have_ws = (ws_size >= (size_t)V * sizeof(float));
    if (have_ws) {
        float* g = (float*)d_ws;
        gumbel_prep_kernel<<<(V + 255) / 256, 256, 0, stream>>>(expn, g, V);
        sampler_kernel<<<T, BLOCK, 0, stream>>>(logits, temps, g, out, V, 0);
    } else {
        sampler_kernel<<<T, BLOCK, 0, stream>>>(logits, temps, expn, out, V, 1);
    }
}